// GNNConv_31774168056050
// MI455X (gfx1250) — compile-verified
//
#include <hip/hip_runtime.h>
#include <hip/hip_bf16.h>
#include <math.h>
#include <stdint.h>

#define CCH  128   // channels
#define TPAD 132   // LDS row pitch (132%64=4 -> conflict-free column reads)
#define ROWS 64    // node rows per block (4 row-tiles of 16)

typedef __attribute__((ext_vector_type(2))) float v2f;
typedef __attribute__((ext_vector_type(8))) float v8f;

// ---------------------------------------------------------------------------
// Async fill of a 64x128 f32 tile (padded pitch TPAD) from global memory.
// Uses gfx1250 GLOBAL_LOAD_ASYNC_TO_LDS_B128 in GVS mode:
//   mem_addr = SADDR(64b, SGPR pair) + VADDR(32b byte offset)
//   LDS dest address (bytes) comes from the VDST VGPR (low 32 bits of the
//   flat shared-aperture address == LDS byte offset per the aperture rules).
// Completed with s_wait_asynccnt 0; caller must __syncthreads() after.
// ---------------------------------------------------------------------------
__device__ __forceinline__ void async_tile_load(float (*buf)[TPAD],
                                                const float* __restrict__ g,
                                                int row0, int N, int tid) {
    unsigned lds_base = (unsigned)(uintptr_t)&buf[0][0];
    for (int i = tid; i < ROWS * (CCH / 4); i += 256) {
        int r  = i >> 5;              // row within tile
        int c4 = (i & 31) << 2;       // starting column (float4 chunk)
        int gr = row0 + r; if (gr >= N) gr = N - 1;   // clamp (guarded stores later)
        unsigned goff  = ((unsigned)gr * CCH + (unsigned)c4) * 4u;
        unsigned laddr = lds_base + ((unsigned)r * TPAD + (unsigned)c4) * 4u;
        asm volatile("global_load_async_to_lds_b128 %0, %1, %2"
                     :: "v"(laddr), "v"(goff), "s"(g) : "memory");
    }
    asm volatile("s_wait_asynccnt 0x0" ::: "memory");
}

// ---------------------------------------------------------------------------
// Kernel 1: zero the aggregation buffer
// ---------------------------------------------------------------------------
__global__ void zero_kernel(float* __restrict__ p, size_t n) {
    size_t i = (size_t)blockIdx.x * blockDim.x + threadIdx.x;
    if (i < n) p[i] = 0.0f;
}

// ---------------------------------------------------------------------------
// Kernel 2: fused node MLP
//   hidden = relu(x @ h_w1 + h_b1)         (LDS, reuses the x-tile buffer)
//   delta  = tanh(hidden @ h_w2 + h_b2)    -> global [N,3]
//   xf     = x @ f_w[3:131] + f_b          -> global [N,128]
// Block = 64 rows; wave w owns a 16-col tile; per k-step one B fragment
// feeds 4 row-tile WMMAs (V_WMMA_F32_16X16X4_F32).
// ---------------------------------------------------------------------------
__global__ void __launch_bounds__(256)
node_mlp_kernel(const float* __restrict__ x,
                const float* __restrict__ h_w1, const float* __restrict__ h_b1,
                const float* __restrict__ h_w2, const float* __restrict__ h_b2,
                const float* __restrict__ f_w,  const float* __restrict__ f_b,
                float* __restrict__ xf, float* __restrict__ delta, int N) {
    __shared__ float buf[ROWS][TPAD];     // x tile, then hidden tile

    const int tid  = threadIdx.x;
    const int wave = tid >> 5;            // 0..7 -> column tile
    const int lane = tid & 31;
    const int row0 = blockIdx.x * ROWS;

    async_tile_load(buf, x, row0, N, tid);
    __syncthreads();

    const int half = lane >> 4;           // 0: lanes 0-15, 1: lanes 16-31
    const int r    = lane & 15;
    const int n0   = wave * 16;
    const float* fw3 = f_w + 3 * CCH;     // rows 3..130 of f_w ([131,128])

    v8f acc_h[4] = {};
    v8f acc_f[4] = {};
    for (int k = 0; k < CCH; k += 4) {
        const int ka = k + 2 * half;
        v2f bh; bh.x = h_w1[(size_t)ka * CCH + n0 + r];
                bh.y = h_w1[(size_t)(ka + 1) * CCH + n0 + r];
        v2f bf; bf.x = fw3[(size_t)ka * CCH + n0 + r];
                bf.y = fw3[(size_t)(ka + 1) * CCH + n0 + r];
        #pragma unroll
        for (int t = 0; t < 4; t++) {
            v2f a; a.x = buf[t * 16 + r][ka];
                   a.y = buf[t * 16 + r][ka + 1];
            acc_h[t] = __builtin_amdgcn_wmma_f32_16x16x4_f32(false, a, false, bh,
                                                             (short)0, acc_h[t], false, false);
            acc_f[t] = __builtin_amdgcn_wmma_f32_16x16x4_f32(false, a, false, bf,
                                                             (short)0, acc_f[t], false, false);
        }
    }
    __syncthreads();   // everyone done reading the x tile -> safe to overwrite

    const float b1  = h_b1[n0 + r];
    const float bfb = f_b[n0 + r];
    #pragma unroll
    for (int t = 0; t < 4; t++) {
        #pragma unroll
        for (int i = 0; i < 8; i++) {
            const int m = t * 16 + i + 8 * half;        // row within tile
            float h = acc_h[t][i] + b1;
            buf[m][n0 + r] = h > 0.0f ? h : 0.0f;        // hidden
            if (row0 + m < N)
                xf[(size_t)(row0 + m) * CCH + n0 + r] = acc_f[t][i] + bfb;
        }
    }
    __syncthreads();

    // tiny head: delta = tanh(hidden @ h_w2 + h_b2), h_w2 is [128,3]
    if (tid < ROWS * 3) {
        const int m = tid / 3, c = tid % 3;
        float acc = h_b2[c];
        for (int k = 0; k < CCH; k++) acc += buf[m][k] * h_w2[k * 3 + c];
        if (row0 + m < N)
            delta[(size_t)(row0 + m) * 3 + c] = tanhf(acc);
    }
}

// ---------------------------------------------------------------------------
// Kernel 3: per-edge message + scatter-add
//   m = relu( (pos_j - pos_i + delta_i) @ f_w[0:3] + xf[j] )
//   aggr[i] += m       (global_atomic_add_f32; xf/aggr are L2-resident)
// One wave per edge (32 lanes x 4 channels). f_w[0:3] staged in LDS.
// ---------------------------------------------------------------------------
__global__ void __launch_bounds__(256)
edge_kernel(const float* __restrict__ pos,
            const int*   __restrict__ eidx,    // [2,E] int32
            const float* __restrict__ xf,
            const float* __restrict__ delta,
            const float* __restrict__ f_w,
            float* __restrict__ aggr, int E) {
    __shared__ float fwg[3][CCH];
    const int tid = threadIdx.x;
    for (int i = tid; i < 3 * CCH; i += 256)
        fwg[i >> 7][i & (CCH - 1)] = f_w[i];
    __syncthreads();

    const int wave = tid >> 5;
    const int lane = tid & 31;
    const long long e = (long long)blockIdx.x * 8 + wave;
    if (e >= E) return;

    const int j = eidx[e];                 // src
    const int i = eidx[(size_t)E + e];     // dst

    const float gx = pos[(size_t)j * 3 + 0] - pos[(size_t)i * 3 + 0] + delta[(size_t)i * 3 + 0];
    const float gy = pos[(size_t)j * 3 + 1] - pos[(size_t)i * 3 + 1] + delta[(size_t)i * 3 + 1];
    const float gz = pos[(size_t)j * 3 + 2] - pos[(size_t)i * 3 + 2] + delta[(size_t)i * 3 + 2];

    const int c0 = lane * 4;
    const float4 v  = *(const float4*)(xf + (size_t)j * CCH + c0);   // global_load_b128
    const float4 w0 = *(const float4*)(&fwg[0][c0]);
    const float4 w1 = *(const float4*)(&fwg[1][c0]);
    const float4 w2 = *(const float4*)(&fwg[2][c0]);

    float m0 = fmaf(gx, w0.x, fmaf(gy, w1.x, fmaf(gz, w2.x, v.x)));
    float m1 = fmaf(gx, w0.y, fmaf(gy, w1.y, fmaf(gz, w2.y, v.y)));
    float m2 = fmaf(gx, w0.z, fmaf(gy, w1.z, fmaf(gz, w2.z, v.z)));
    float m3 = fmaf(gx, w0.w, fmaf(gy, w1.w, fmaf(gz, w2.w, v.w)));
    m0 = m0 > 0.0f ? m0 : 0.0f;
    m1 = m1 > 0.0f ? m1 : 0.0f;
    m2 = m2 > 0.0f ? m2 : 0.0f;
    m3 = m3 > 0.0f ? m3 : 0.0f;

    float* dst = aggr + (size_t)i * CCH + c0;
    atomicAdd(dst + 0, m0);
    atomicAdd(dst + 1, m1);
    atomicAdd(dst + 2, m2);
    atomicAdd(dst + 3, m3);
}

// ---------------------------------------------------------------------------
// Kernel 4: node update + residual
//   out = x + relu(aggr @ g_w1 + g_b1) @ g_w2 + g_b2
// Two chained WMMA GEMMs through one reused LDS tile.
// ---------------------------------------------------------------------------
__global__ void __launch_bounds__(256)
node_out_kernel(const float* __restrict__ x,
                const float* __restrict__ aggr,
                const float* __restrict__ g_w1, const float* __restrict__ g_b1,
                const float* __restrict__ g_w2, const float* __restrict__ g_b2,
                float* __restrict__ out, int N) {
    __shared__ float buf[ROWS][TPAD];     // aggr tile, then t tile

    const int tid  = threadIdx.x;
    const int wave = tid >> 5;
    const int lane = tid & 31;
    const int row0 = blockIdx.x * ROWS;

    async_tile_load(buf, aggr, row0, N, tid);
    __syncthreads();

    const int half = lane >> 4;
    const int r    = lane & 15;
    const int n0   = wave * 16;

    // GEMM 1: t = relu(aggr @ g_w1 + g_b1)
    v8f acc1[4] = {};
    for (int k = 0; k < CCH; k += 4) {
        const int ka = k + 2 * half;
        v2f b; b.x = g_w1[(size_t)ka * CCH + n0 + r];
               b.y = g_w1[(size_t)(ka + 1) * CCH + n0 + r];
        #pragma unroll
        for (int t = 0; t < 4; t++) {
            v2f a; a.x = buf[t * 16 + r][ka];
                   a.y = buf[t * 16 + r][ka + 1];
            acc1[t] = __builtin_amdgcn_wmma_f32_16x16x4_f32(false, a, false, b,
                                                            (short)0, acc1[t], false, false);
        }
    }
    __syncthreads();   // done reading aggr tile

    const float b1 = g_b1[n0 + r];
    #pragma unroll
    for (int t = 0; t < 4; t++) {
        #pragma unroll
        for (int i = 0; i < 8; i++) {
            const int m = t * 16 + i + 8 * half;
            float v = acc1[t][i] + b1;
            buf[m][n0 + r] = v > 0.0f ? v : 0.0f;
        }
    }
    __syncthreads();

    // GEMM 2: o = t @ g_w2 + g_b2 + x
    v8f acc2[4] = {};
    for (int k = 0; k < CCH; k += 4) {
        const int ka = k + 2 * half;
        v2f b; b.x = g_w2[(size_t)ka * CCH + n0 + r];
               b.y = g_w2[(size_t)(ka + 1) * CCH + n0 + r];
        #pragma unroll
        for (int t = 0; t < 4; t++) {
            v2f a; a.x = buf[t * 16 + r][ka];
                   a.y = buf[t * 16 + r][ka + 1];
            acc2[t] = __builtin_amdgcn_wmma_f32_16x16x4_f32(false, a, false, b,
                                                            (short)0, acc2[t], false, false);
        }
    }
    const float b2 = g_b2[n0 + r];
    #pragma unroll
    for (int t = 0; t < 4; t++) {
        #pragma unroll
        for (int i = 0; i < 8; i++) {
            const int m = t * 16 + i + 8 * half;
            if (row0 + m < N) {
                const size_t idx = (size_t)(row0 + m) * CCH + n0 + r;
                out[idx] = x[idx] + acc2[t][i] + b2;
            }
        }
    }
}

// ---------------------------------------------------------------------------
extern "C" void kernel_launch(void* const* d_in, const int* in_sizes, int n_in,
                              void* d_out, int out_size, void* d_ws, size_t ws_size,
                              hipStream_t stream) {
    const float* x     = (const float*)d_in[0];
    const float* pos   = (const float*)d_in[1];
    const int*   eidx  = (const int*)d_in[2];
    const float* h_w1  = (const float*)d_in[3];
    const float* h_b1  = (const float*)d_in[4];
    const float* h_w2  = (const float*)d_in[5];
    const float* h_b2  = (const float*)d_in[6];
    const float* f_w   = (const float*)d_in[7];
    const float* f_b   = (const float*)d_in[8];
    const float* g_w1  = (const float*)d_in[9];
    const float* g_b1  = (const float*)d_in[10];
    const float* g_w2  = (const float*)d_in[11];
    const float* g_b2  = (const float*)d_in[12];
    float* out = (float*)d_out;

    const int N = in_sizes[0] / CCH;   // 50000
    const int E = in_sizes[2] / 2;     // 600000

    // workspace layout (floats), segments kept 256B-aligned
    float*  xf       = (float*)d_ws;
    float*  delta    = xf + (size_t)N * CCH;
    size_t  delta_sz = (((size_t)N * 3) + 63) & ~(size_t)63;
    float*  aggr     = delta + delta_sz;

    {   // 1) zero aggr
        size_t n = (size_t)N * CCH;
        int blocks = (int)((n + 1023) / 1024);
        zero_kernel<<<blocks, 1024, 0, stream>>>(aggr, n);
    }
    {   // 2) node MLP: delta + xf
        int blocks = (N + ROWS - 1) / ROWS;
        node_mlp_kernel<<<blocks, 256, 0, stream>>>(
            x, h_w1, h_b1, h_w2, h_b2, f_w, f_b, xf, delta, N);
    }
    {   // 3) edge messages + scatter-add
        int blocks = (E + 7) / 8;
        edge_kernel<<<blocks, 256, 0, stream>>>(pos, eidx, xf, delta, f_w, aggr, E);
    }
    {   // 4) node update + residual
        int blocks = (N + ROWS - 1) / ROWS;
        node_out_kernel<<<blocks, 256, 0, stream>>>(
            x, aggr, g_w1, g_b1, g_w2, g_b2, out, N);
    }
}